// SecondaryPredictor_42923903156345
// MI455X (gfx1250) — compile-verified
//
#include <hip/hip_runtime.h>

typedef __attribute__((ext_vector_type(2))) float v2f;
typedef __attribute__((ext_vector_type(8))) float v8f;

#define LSEQ 1024
#define DMOD 512
#define HDIM 256

// ---------------------------------------------------------------------------
// Phase 1: P[l, 0:256]   = seq @ W1[:, :512].T + b1      (pa' = pa + b1)
//          P[l, 256:512] = seq @ W1[:, 512:].T           (pb)
// One wave per 16x16 output tile, V_WMMA_F32_16X16X4_F32 over K=512.
// ---------------------------------------------------------------------------
__global__ __launch_bounds__(32)
void sp_gemm_wmma(const float* __restrict__ A,    // [1024,512] seq_feats
                  const float* __restrict__ W1,   // [256,1024]
                  const float* __restrict__ b1,   // [256]
                  float* __restrict__ P)          // [1024,512]
{
    const int lane   = threadIdx.x;        // 0..31 (wave32)
    const int lane15 = lane & 15;
    const int koff   = (lane >> 4) << 1;   // lanes 0-15 -> K=0,1 ; lanes 16-31 -> K=2,3
    const int n0     = blockIdx.x * 16;    // output col tile (0..511)
    const int m0     = blockIdx.y * 16;    // output row tile (0..1023)

    const int n = n0 + lane15;
    // B[k,n] = (n<256) ? W1[n, k] : W1[n-256, 512+k]  (W1 row-major, ld=1024)
    const float* rowA = A + (m0 + lane15) * DMOD;
    const float* rowB = (n < HDIM) ? (W1 + n * (2 * DMOD))
                                   : (W1 + (n - HDIM) * (2 * DMOD) + DMOD);

    v8f c = {0.f, 0.f, 0.f, 0.f, 0.f, 0.f, 0.f, 0.f};
    #pragma unroll 8
    for (int k0 = 0; k0 < DMOD; k0 += 4) {
        v2f a = *(const v2f*)(rowA + k0 + koff);   // A[m, k0+koff], A[m, k0+koff+1]
        v2f b = *(const v2f*)(rowB + k0 + koff);   // B[k0+koff, n], B[k0+koff+1, n]
        // (neg_a, A, neg_b, B, c_mod, C, reuse_a, reuse_b)
        c = __builtin_amdgcn_wmma_f32_16x16x4_f32(false, a, false, b,
                                                  (short)0, c, false, false);
    }

    const float bias  = (n < HDIM) ? b1[n] : 0.f;
    const int   mBase = m0 + ((lane >> 4) << 3);   // +8 for upper half-wave
    #pragma unroll
    for (int v = 0; v < 8; ++v)
        P[(mBase + v) * (2 * HDIM) + n] = c[v] + bias;
}

// ---------------------------------------------------------------------------
// Phase 2: for block tile (bi<=bj): both s(i,j) and s(j,i), then fused
// symmetrize + sigmoid. LDS: row-major for wave-broadcast (ii) accesses,
// pitch-33 transposed for lane-indexed (jj) accesses (conflict-free).
// ---------------------------------------------------------------------------
__global__ __launch_bounds__(256)
void sp_pair_scores(const float* __restrict__ P,   // [1024,512]
                    const float* __restrict__ W2,  // [256]
                    const float* __restrict__ b2,  // [1]
                    float* __restrict__ out)       // [1024,1024]
{
    const int bi = blockIdx.y, bj = blockIdx.x;
    if (bj < bi) return;                 // upper triangle incl. diagonal
    const int i0 = bi * 32, j0 = bj * 32;

    __shared__ float s_paI[32 * 256];    // pa'[i0+r][h]   row-major
    __shared__ float s_pbI[32 * 256];    // pb  [i0+r][h]  row-major
    __shared__ float s_paJT[256 * 33];   // pa'[j0+r][h]   transposed, pitch 33
    __shared__ float s_pbJT[256 * 33];   // pb  [j0+r][h]  transposed, pitch 33
    __shared__ float s_w2[256];

    const int tid = threadIdx.x;         // 256 threads
    s_w2[tid] = W2[tid];
    #pragma unroll 4
    for (int r = 0; r < 32; ++r) {
        const int c = tid;               // 0..255, coalesced global reads
        const float aI = P[(i0 + r) * 512 + c];
        const float bI = P[(i0 + r) * 512 + 256 + c];
        const float aJ = P[(j0 + r) * 512 + c];
        const float bJ = P[(j0 + r) * 512 + 256 + c];
        s_paI[r * 256 + c]  = aI;
        s_pbI[r * 256 + c]  = bI;
        s_paJT[c * 33 + r]  = aJ;        // pitch 33: gcd(33,64)=1 -> no conflicts
        s_pbJT[c * 33 + r]  = bJ;
    }
    __syncthreads();

    const int jj  = tid & 31;            // lane -> column, conflict-free jT reads
    const int ii0 = tid >> 5;            // constant per wave -> broadcast I reads
    float accIJ[4] = {0.f, 0.f, 0.f, 0.f};
    float accJI[4] = {0.f, 0.f, 0.f, 0.f};

    #pragma unroll 2
    for (int h = 0; h < 256; ++h) {
        const float bJ = s_pbJT[h * 33 + jj];
        const float aJ = s_paJT[h * 33 + jj];
        const float w  = s_w2[h];
        #pragma unroll
        for (int r = 0; r < 4; ++r) {
            const int ii = ii0 + (r << 3);
            const float aI = s_paI[ii * 256 + h];   // broadcast
            const float bI = s_pbI[ii * 256 + h];   // broadcast
            accIJ[r] = fmaf(fmaxf(aI + bJ, 0.f), w, accIJ[r]);  // s(I,J) term
            accJI[r] = fmaf(fmaxf(aJ + bI, 0.f), w, accJI[r]);  // s(J,I) term
        }
    }

    const float b2v = b2[0];
    #pragma unroll
    for (int r = 0; r < 4; ++r) {
        const int I = i0 + ii0 + (r << 3);
        const int J = j0 + jj;
        const float sc  = 0.5f * (accIJ[r] + accJI[r]) + b2v;
        const float sig = 1.0f / (1.0f + expf(-sc));
        out[I * LSEQ + J] = sig;
        out[J * LSEQ + I] = sig;   // diagonal blocks: duplicate writes, same value
    }
}

// ---------------------------------------------------------------------------
extern "C" void kernel_launch(void* const* d_in, const int* in_sizes, int n_in,
                              void* d_out, int out_size, void* d_ws, size_t ws_size,
                              hipStream_t stream) {
    const float* seq = (const float*)d_in[0];  // [1024,512]
    const float* W1  = (const float*)d_in[1];  // [256,1024]
    const float* b1  = (const float*)d_in[2];  // [256]
    const float* W2  = (const float*)d_in[3];  // [1,256]
    const float* b2  = (const float*)d_in[4];  // [1]
    float* out = (float*)d_out;                // [1024,1024]
    float* P   = (float*)d_ws;                 // 1024*512 floats = 2 MB scratch

    sp_gemm_wmma<<<dim3(32, 64), 32, 0, stream>>>(seq, W1, b1, P);
    sp_pair_scores<<<dim3(32, 32), 256, 0, stream>>>(P, W2, b2, out);
}